// HereroCatPredictor_8332236554763
// MI455X (gfx1250) — compile-verified
//
#include <hip/hip_runtime.h>

typedef __attribute__((ext_vector_type(2))) float v2f;
typedef __attribute__((ext_vector_type(8))) float v8f;

// ---------------------------------------------------------------------------
// Pass 1: per-node inverse L2 norm of concat([x_row, h_row]) (128 floats).
// One wave32 per node; lane L<16 reads x[row][4L..4L+3], lane L>=16 reads
// h[row][4(L-16)..]. Coalesced float4 loads, shfl_xor butterfly reduction.
// ---------------------------------------------------------------------------
__global__ __launch_bounds__(256) void node_invnorm_kernel(
    const float* __restrict__ x, const float* __restrict__ h,
    float* __restrict__ invn, int n_nodes) {
  int gtid = blockIdx.x * blockDim.x + threadIdx.x;
  int node = gtid >> 5;
  int lane = threadIdx.x & 31;
  if (node >= n_nodes) return;

  const float* p = (lane < 16)
      ? (x + (size_t)node * 64 + 4 * lane)
      : (h + (size_t)node * 64 + 4 * (lane - 16));
  float4 v = *(const float4*)p;
  float s = v.x * v.x + v.y * v.y + v.z * v.z + v.w * v.w;
#pragma unroll
  for (int off = 16; off > 0; off >>= 1) s += __shfl_xor(s, off, 32);
  if (lane == 0) invn[node] = 1.0f / __builtin_sqrtf(s);
}

// ---------------------------------------------------------------------------
// Pass 2: 16 edges per wave via V_WMMA_F32_16X16X4_F32.
// A (16x4 f32): lane L holds row M = L&15, K-pair 2*(L>>4)   -> b64 gather
// B (4x16 f32): lane L holds col N = L&15, K-pair 2*(L>>4)   -> b64 gather
// 32 chained WMMAs accumulate the full K=128 dot; edge i's dot is C[i][i]:
//   i in 0..7  -> lane i,      accumulator VGPR i
//   i in 8..15 -> lane 16+i,   accumulator VGPR i-8
// Writing lanes already hold src/dst of their own edge (m == edge-e0),
// so no index reload is needed at writeback.
// ---------------------------------------------------------------------------
__global__ __launch_bounds__(256) void edge_cosine_wmma_kernel(
    const float* __restrict__ xd, const float* __restrict__ hd,
    const float* __restrict__ xg, const float* __restrict__ hg,
    const int* __restrict__ src, const int* __restrict__ dst,
    const float* __restrict__ inv_nd, const float* __restrict__ inv_ng,
    float* __restrict__ out, int n_edges) {
  int gtid = blockIdx.x * blockDim.x + threadIdx.x;
  int wave = gtid >> 5;
  int lane = threadIdx.x & 31;
  int e0 = wave * 16;
  if (e0 >= n_edges) return;  // uniform per wave

  int m  = lane & 15;          // row (A) / col (B) owned by this lane
  int kk = (lane >> 4) * 2;    // K-pair offset within each 4-wide K chunk

  int ei = e0 + m;
  if (ei > n_edges - 1) ei = n_edges - 1;  // clamp tail (EXEC must stay full)
  int sRow = src[ei];
  int dRow = dst[ei];

  const float* ax = xd + (size_t)sRow * 64 + kk;
  const float* ah = hd + (size_t)sRow * 64 + kk;
  const float* bx = xg + (size_t)dRow * 64 + kk;
  const float* bh = hg + (size_t)dRow * 64 + kk;

  v8f c = {};
  // x-half of the concat feature (columns 0..63)
#pragma unroll
  for (int k0 = 0; k0 < 64; k0 += 4) {
    v2f a = *(const v2f*)(ax + k0);
    v2f b = *(const v2f*)(bx + k0);
    c = __builtin_amdgcn_wmma_f32_16x16x4_f32(false, a, false, b,
                                              (short)0, c, false, false);
  }
  // h-half of the concat feature (columns 64..127)
#pragma unroll
  for (int k0 = 0; k0 < 64; k0 += 4) {
    v2f a = *(const v2f*)(ah + k0);
    v2f b = *(const v2f*)(bh + k0);
    c = __builtin_amdgcn_wmma_f32_16x16x4_f32(false, a, false, b,
                                              (short)0, c, false, false);
  }

  // Extract diagonal, scale by inverse-norm product, store.
  bool low  = lane < 8;    // owns diag 0..7   (VGPR = lane)
  bool high = lane >= 24;  // owns diag 8..15  (VGPR = lane-24)
  if (low || high) {
    int di   = low ? lane : (lane - 24);
    int edge = e0 + m;     // == ei (un-clamped guard below)
    if (edge < n_edges) {
      float diag = c[0];
      if (di == 1) diag = c[1];
      if (di == 2) diag = c[2];
      if (di == 3) diag = c[3];
      if (di == 4) diag = c[4];
      if (di == 5) diag = c[5];
      if (di == 6) diag = c[6];
      if (di == 7) diag = c[7];
      // This lane's sRow/dRow are exactly src[edge]/dst[edge].
      out[edge] = diag * inv_nd[sRow] * inv_ng[dRow];
    }
  }
}

extern "C" void kernel_launch(void* const* d_in, const int* in_sizes, int n_in,
                              void* d_out, int out_size, void* d_ws, size_t ws_size,
                              hipStream_t stream) {
  const float* xd = (const float*)d_in[0];
  const float* hd = (const float*)d_in[1];
  const float* xg = (const float*)d_in[2];
  const float* hg = (const float*)d_in[3];
  const int*  src = (const int*)d_in[4];
  const int*  dst = (const int*)d_in[5];

  int n_d = in_sizes[0] / 64;
  int n_g = in_sizes[2] / 64;
  int E   = in_sizes[4];

  float* inv_nd = (float*)d_ws;
  float* inv_ng = inv_nd + n_d;
  float* out = (float*)d_out;

  const int TPB = 256;  // 8 wave32s per block

  // Pass 1: inverse norms (one wave per node).
  {
    long threads = (long)n_d * 32;
    node_invnorm_kernel<<<(int)((threads + TPB - 1) / TPB), TPB, 0, stream>>>(
        xd, hd, inv_nd, n_d);
  }
  {
    long threads = (long)n_g * 32;
    node_invnorm_kernel<<<(int)((threads + TPB - 1) / TPB), TPB, 0, stream>>>(
        xg, hg, inv_ng, n_g);
  }

  // Pass 2: 16 edges per wave, WMMA dot + normalize.
  {
    long waves   = (E + 15) / 16;
    long threads = waves * 32;
    edge_cosine_wmma_kernel<<<(int)((threads + TPB - 1) / TPB), TPB, 0, stream>>>(
        xd, hd, xg, hg, src, dst, inv_nd, inv_ng, out, E);
  }
}